// mAP_77721728188847
// MI455X (gfx1250) — compile-verified
//
#include <hip/hip_runtime.h>
#include <stdint.h>

#define N_PRED 20000
#define N_SORT 32768   // next pow2 >= N_PRED for bitonic sort
#define G_GT   1000
#define NEG_INF_F (-1e30f)

typedef __attribute__((ext_vector_type(2))) float        v2f;
typedef __attribute__((ext_vector_type(8))) float        v8f;
typedef __attribute__((ext_vector_type(4))) unsigned int v4u;
typedef __attribute__((ext_vector_type(4))) int          v4i;
typedef __attribute__((ext_vector_type(8))) int          v8i;

// ---------------------------------------------------------------------------
// K1: build 64-bit sort keys per class.
// key = (order-preserving score bits) << 32 | (0xFFFFFFFF - index)
// Descending sort on key == descending score, ties -> smaller original index
// (matches stable jnp.argsort(-scores)). Padding entries get key 0 (sort last).
// ---------------------------------------------------------------------------
__global__ void build_keys_kernel(const float* __restrict__ probits,
                                  const int* __restrict__ labels,
                                  unsigned long long* __restrict__ keys) {
  int ci = blockIdx.y;          // class index 0..2 -> class c = ci+1
  int c  = ci + 1;
  int j  = blockIdx.x * blockDim.x + threadIdx.x;
  if (j >= N_SORT) return;
  unsigned long long key = 0ull;
  if (j < N_PRED) {
    bool valid = (labels[j] == c);
    float s = valid ? probits[j] : NEG_INF_F;
    unsigned b = __float_as_uint(s);
    unsigned ord = (b & 0x80000000u) ? ~b : (b | 0x80000000u);
    key = ((unsigned long long)ord << 32) |
          (unsigned long long)(0xFFFFFFFFu - (unsigned)j);
  }
  keys[(size_t)ci * N_SORT + j] = key;
}

// ---------------------------------------------------------------------------
// K2: one bitonic compare-exchange pass (descending order), all 3 classes in y.
// ---------------------------------------------------------------------------
__global__ void bitonic_pass_kernel(unsigned long long* __restrict__ keys,
                                    int k, int j) {
  int ci = blockIdx.y;
  unsigned i = blockIdx.x * blockDim.x + threadIdx.x;
  unsigned ixj = i ^ (unsigned)j;
  if (i >= N_SORT || ixj <= i || ixj >= N_SORT) return;
  unsigned long long* base = keys + (size_t)ci * N_SORT;
  unsigned long long a = base[i];
  unsigned long long b = base[ixj];
  bool desc = ((i & (unsigned)k) == 0);
  bool do_swap = desc ? (a < b) : (a > b);
  if (do_swap) { base[i] = b; base[ixj] = a; }
}

// ---------------------------------------------------------------------------
// K3: per sorted prediction, best IoU + best GT over all 1000 GT boxes.
// One wave = one 16-pred tile (WMMA column dim). Loops over 2x16-GT tiles.
// WMMA f32 16x16x4 computes denom = area1[g] + area2[n] - inter as
// D = A(16x4) x B(4x16) + C with A=[area1|1|0|0], B=[1;area2;0;0], C=-inter.
// GT tensor (1000x5 f32, 20KB) is DMA'd into LDS via the Tensor Data Mover.
// ---------------------------------------------------------------------------
__global__ __launch_bounds__(256) void best_iou_kernel(
    const float* __restrict__ pred_boxes,          // [N,4] t,l,b,r
    const int* __restrict__ labels,                // [N]
    const float* __restrict__ gt,                  // [G,5] label,t,l,b,r
    const unsigned long long* __restrict__ keys,   // [3][N_SORT]
    float* __restrict__ best_iou,                  // [3][N]
    int* __restrict__ best_gv) {                   // [3][N] (gt idx | valid<<30)
  __shared__ float s_gt[G_GT * 5];

  int ci = blockIdx.y;
  int c  = ci + 1;

  // ---- stage GT into LDS via TDM (gfx1250 tensor DMA) ----
  if (threadIdx.x == 0) {
    unsigned lds_addr = (unsigned)(uintptr_t)(&s_gt[0]);
    unsigned long long ga = (unsigned long long)(uintptr_t)gt;
    const int elems = G_GT * 5;  // 5000 x f32
    v4u g0;
    g0.x = 1u;                                           // count=1, user desc
    g0.y = lds_addr;                                     // lds_addr (bytes)
    g0.z = (unsigned)(ga & 0xFFFFFFFFull);               // global_addr[31:0]
    g0.w = (unsigned)((ga >> 32) & 0x01FFFFFFull) |
           0x80000000u;                                  // addr[56:32], type=2
    v8i g1;
    g1[0] = (2 << 16);                                   // data_size = 4B
    g1[1] = (elems & 0xFFFF) << 16;                      // tensor_dim0 lo
    g1[2] = ((elems >> 16) & 0xFFFF) | (1 << 16);        // dim0 hi, dim1=1
    g1[3] = ((elems & 0xFFFF) << 16);                    // tile_dim0 = 5000
    g1[4] = 1;                                           // tile_dim1=1
    g1[5] = elems;                                       // dim0_stride lo
    g1[6] = 0;
    g1[7] = 0;
    v4i g2 = {0, 0, 0, 0};
    v4i g3 = {0, 0, 0, 0};
#if defined(__clang_major__) && __clang_major__ >= 23
    v8i g4 = {0, 0, 0, 0, 0, 0, 0, 0};
    __builtin_amdgcn_tensor_load_to_lds(g0, g1, g2, g3, g4, 0);
#else
    __builtin_amdgcn_tensor_load_to_lds(g0, g1, g2, g3, 0);
#endif
    __builtin_amdgcn_s_wait_tensorcnt(0);
  }
  __syncthreads();

  int lane = threadIdx.x & 31;
  int wave = threadIdx.x >> 5;
  int tile = blockIdx.x * 8 + wave;   // 16-pred tile
  int n    = lane & 15;               // column within tile
  int r    = tile * 16 + n;           // sorted rank
  bool in_range = (r < N_PRED);
  bool hi = (lane >= 16);             // upper half: rows m+8 of C layout

  unsigned long long key = keys[(size_t)ci * N_SORT + (in_range ? r : 0)];
  unsigned j = 0xFFFFFFFFu - (unsigned)(key & 0xFFFFFFFFull);
  if (!in_range) j = 0;
  bool pvalid = in_range && (labels[j] == c);

  float pt = pred_boxes[j * 4 + 0];
  float pl = pred_boxes[j * 4 + 1];
  float pb = pred_boxes[j * 4 + 2];
  float pr = pred_boxes[j * 4 + 3];
  float parea = (pb - pt) * (pr - pl);

  // B operand: lanes 0-15 carry rows K0,K1 = (1, area2[n]); lanes 16-31 = K2,K3 = 0
  v2f bmat;
  bmat.x = hi ? 0.0f : 1.0f;
  bmat.y = hi ? 0.0f : parea;

  float bestv = -2.0f;
  int bestm = 0;

  // one 16-GT (M) tile: inter per lane in C layout, denom via WMMA, argmax fold
  auto process_tile = [&](int gbase) {
    int am = gbase + lane;
    int ams = (am < G_GT) ? am : 0;
    float a_area = (s_gt[ams * 5 + 3] - s_gt[ams * 5 + 1]) *
                   (s_gt[ams * 5 + 4] - s_gt[ams * 5 + 2]);
    bool arow_ok = (!hi) && (am < G_GT);
    v2f amat;
    amat.x = arow_ok ? a_area : 0.0f;
    amat.y = arow_ok ? 1.0f : 0.0f;

    v8f cacc;
    float inter[8];
    bool gok[8];
#pragma unroll
    for (int rr = 0; rr < 8; ++rr) {
      int m = gbase + rr + (hi ? 8 : 0);
      int ms = (m < G_GT) ? m : 0;
      float g_t = s_gt[ms * 5 + 1];
      float g_l = s_gt[ms * 5 + 2];
      float g_b = s_gt[ms * 5 + 3];
      float g_r = s_gt[ms * 5 + 4];
      float ih = fmaxf(fminf(pb, g_b) - fmaxf(pt, g_t), 0.0f);
      float iw = fmaxf(fminf(pr, g_r) - fmaxf(pl, g_l), 0.0f);
      float itv = ih * iw;
      inter[rr] = itv;
      gok[rr] = (m < G_GT) && (s_gt[ms * 5 + 0] == (float)c);
      cacc[rr] = -itv;
    }

    // D[m][n] = area1[m] + area2[n] - inter[m][n]
    v8f d = __builtin_amdgcn_wmma_f32_16x16x4_f32(
        false, amat, false, bmat, (short)0, cacc, false, false);

#pragma unroll
    for (int rr = 0; rr < 8; ++rr) {
      int m = gbase + rr + (hi ? 8 : 0);
      float q = inter[rr] / d[rr];
      float iou = gok[rr] ? q : -1.0f;
      if (iou > bestv) { bestv = iou; bestm = m; }  // strict >: first-max wins
    }
  };

  for (int gbase = 0; gbase < G_GT; gbase += 32) {
    process_tile(gbase);        // GT rows gbase   .. gbase+15
    process_tile(gbase + 16);   // GT rows gbase+16.. gbase+31 (clamped)
  }

  // combine the two row-halves (same column n) held by lane and lane^16
  float ov = __shfl_xor(bestv, 16, 32);
  int   om = __shfl_xor(bestm, 16, 32);
  if (ov > bestv || (ov == bestv && om < bestm)) { bestv = ov; bestm = om; }

  if (!hi && in_range) {
    best_iou[(size_t)ci * N_PRED + r] = bestv;
    best_gv[(size_t)ci * N_PRED + r] = bestm | (pvalid ? (1 << 30) : 0);
  }
}

// ---------------------------------------------------------------------------
// K4: sequential greedy matching + 11-point interpolated AP, one block/class.
// Chunks of (best_iou, best_gv) are staged into LDS cooperatively; lane 0
// runs the serial matcher (recall is monotone -> running max per threshold).
// ---------------------------------------------------------------------------
__global__ __launch_bounds__(256) void match_ap_kernel(
    const float* __restrict__ gt,
    const float* __restrict__ best_iou,
    const int* __restrict__ best_gv,
    float* __restrict__ ap_out) {
  __shared__ int   matched[G_GT];
  __shared__ float s_bi[2048];
  __shared__ int   s_gv[2048];
  __shared__ int   s_numgt;

  int ci = blockIdx.x;
  int c  = ci + 1;

  if (threadIdx.x == 0) s_numgt = 0;
  for (int i = threadIdx.x; i < G_GT; i += blockDim.x) matched[i] = 0;
  __syncthreads();
  int cnt = 0;
  for (int i = threadIdx.x; i < G_GT; i += blockDim.x)
    if (gt[i * 5] == (float)c) cnt++;
  atomicAdd(&s_numgt, cnt);
  __syncthreads();
  float numgt = (float)s_numgt;

  float tp = 0.0f, fp = 0.0f;
  float pmax[11];
#pragma unroll
  for (int k = 0; k < 11; ++k) pmax[k] = 0.0f;
  int anyv = 0;

  for (int base = 0; base < N_PRED; base += 2048) {
    int count = min(2048, N_PRED - base);
    __syncthreads();
    for (int i = threadIdx.x; i < count; i += blockDim.x) {
      s_bi[i] = best_iou[(size_t)ci * N_PRED + base + i];
      s_gv[i] = best_gv[(size_t)ci * N_PRED + base + i];
    }
    __syncthreads();
    if (threadIdx.x == 0) {
      for (int i = 0; i < count; ++i) {
        int gv = s_gv[i];
        int v  = (gv >> 30) & 1;
        int bg = gv & 0x3FFFFFFF;
        float bi = s_bi[i];
        int is_tp = v && (bi > 0.5f) && (matched[bg] == 0);
        if (is_tp) { matched[bg] = 1; tp += 1.0f; }
        else if (v) fp += 1.0f;
        anyv |= v;
        float prec = tp / (tp + fp + 1e-5f);
        float rec  = tp / (numgt + 1e-5f);
#pragma unroll
        for (int k = 0; k < 11; ++k)
          if (rec >= 0.1f * (float)k) pmax[k] = fmaxf(pmax[k], prec);
      }
    }
  }
  if (threadIdx.x == 0) {
    float s = 0.0f;
#pragma unroll
    for (int k = 0; k < 11; ++k) s += pmax[k];
    float ap = s / 11.0f;
    if (!(anyv && s_numgt > 0)) ap = 0.0f;
    ap_out[ci] = ap;
  }
}

// K5: mean over classes -> scalar output
__global__ void finalize_kernel(const float* __restrict__ ap,
                                float* __restrict__ out) {
  if (blockIdx.x == 0 && threadIdx.x == 0)
    out[0] = (ap[0] + ap[1] + ap[2]) * (1.0f / 3.0f);
}

extern "C" void kernel_launch(void* const* d_in, const int* in_sizes, int n_in,
                              void* d_out, int out_size, void* d_ws,
                              size_t ws_size, hipStream_t stream) {
  (void)in_sizes; (void)n_in; (void)out_size; (void)ws_size;
  const float* probits = (const float*)d_in[0];
  const float* pboxes  = (const float*)d_in[1];
  const int*   plabels = (const int*)d_in[2];
  const float* gt      = (const float*)d_in[3];
  float* out = (float*)d_out;

  char* ws = (char*)d_ws;
  unsigned long long* keys = (unsigned long long*)ws;      // 3*32768*8 = 768KB
  size_t off = 3ull * N_SORT * sizeof(unsigned long long);
  float* best_iou = (float*)(ws + off); off += 3ull * N_PRED * sizeof(float);
  int*   best_gv  = (int*)(ws + off);   off += 3ull * N_PRED * sizeof(int);
  float* ap       = (float*)(ws + off);

  dim3 gk1((N_SORT + 255) / 256, 3);
  build_keys_kernel<<<gk1, 256, 0, stream>>>(probits, plabels, keys);

  dim3 gk2(N_SORT / 256, 3);
  for (int k = 2; k <= N_SORT; k <<= 1)
    for (int j = k >> 1; j >= 1; j >>= 1)
      bitonic_pass_kernel<<<gk2, 256, 0, stream>>>(keys, k, j);

  dim3 gk3((N_PRED / 16 + 7) / 8, 3);   // 157 blocks x 3 classes
  best_iou_kernel<<<gk3, 256, 0, stream>>>(pboxes, plabels, gt, keys,
                                           best_iou, best_gv);

  match_ap_kernel<<<3, 256, 0, stream>>>(gt, best_iou, best_gv, ap);
  finalize_kernel<<<1, 64, 0, stream>>>(ap, out);
}